// IncepGCNBranch_56307021250670
// MI455X (gfx1250) — compile-verified
//
#include <hip/hip_runtime.h>
#include <hip/hip_bf16.h>

#define NNODES 100000
#define NEDGES 1600000
#define FDIM   64
#define BN_EPS 1e-5f

typedef __attribute__((ext_vector_type(2))) float v2f;
typedef __attribute__((ext_vector_type(8))) float v8f;

// ---------------- degree / dinv ----------------

__global__ void zero_f32(float* __restrict__ p, int n) {
    int i = blockIdx.x * blockDim.x + threadIdx.x;
    if (i < n) p[i] = 0.0f;
}

__global__ void deg_count(const int* __restrict__ dst, float* __restrict__ deg) {
    int e = blockIdx.x * blockDim.x + threadIdx.x;
    if (e < NEDGES) atomicAdd(&deg[dst[e]], 1.0f);
}

__global__ void make_dinv(float* __restrict__ deg) {
    int i = blockIdx.x * blockDim.x + threadIdx.x;
    if (i < NNODES) deg[i] = rsqrtf(deg[i] + 1.0f);  // +1 self loop; deg>=1 always
}

// ---------------- BN scale/shift precompute ----------------

__global__ void bn_prep(const float* __restrict__ g, const float* __restrict__ b,
                        const float* __restrict__ m, const float* __restrict__ v,
                        float* __restrict__ scale, float* __restrict__ shift) {
    int f = threadIdx.x;
    if (f < FDIM) {
        float s = g[f] * rsqrtf(v[f] + BN_EPS);
        scale[f] = s;
        shift[f] = b[f] - m[f] * s;
    }
}

// ---------------- fused BN + GEMM via V_WMMA_F32_16X16X4_F32 ----------------
// C[N x 64] = (A .* scale + shift) @ W,  W row-major [64 x 64]
// One wave computes a full 16x64 row block of C (4 accumulators), so the
// BN'd A operand register is reused across 4 WMMAs per K-step.

__global__ void __launch_bounds__(256)
bn_gemm_wmma(const float* __restrict__ A, const float* __restrict__ W,
             const float* __restrict__ scale, const float* __restrict__ shift,
             float* __restrict__ C, int n_rows) {
    const int lane     = threadIdx.x & 31;
    const int wave     = threadIdx.x >> 5;
    const int row_tile = blockIdx.x * (blockDim.x >> 5) + wave;
    if (row_tile * 16 >= n_rows) return;             // uniform across wave

    const int mrow  = row_tile * 16 + (lane & 15);   // A row for this lane
    const int khalf = (lane >> 4) * 2;               // K offset 0 or 2
    const int ncl   = lane & 15;                     // column within a tile

    v8f acc0 = {}, acc1 = {}, acc2 = {}, acc3 = {};
#pragma unroll
    for (int k = 0; k < FDIM; k += 4) {
        const int c0 = k + khalf;                    // even -> float2 aligned
        const float2 av = *(const float2*)(A + (size_t)mrow * FDIM + c0);
        const float2 sv = *(const float2*)(scale + c0);
        const float2 tv = *(const float2*)(shift + c0);
        v2f a;
        a.x = av.x * sv.x + tv.x;
        a.y = av.y * sv.y + tv.y;

        const float* Wr0 = W + (size_t)c0 * FDIM + ncl;   // row c0
        const float* Wr1 = Wr0 + FDIM;                    // row c0+1
        v2f b0, b1, b2, b3;
        b0.x = Wr0[0];  b0.y = Wr1[0];
        b1.x = Wr0[16]; b1.y = Wr1[16];
        b2.x = Wr0[32]; b2.y = Wr1[32];
        b3.x = Wr0[48]; b3.y = Wr1[48];

        // (neg_a, A, neg_b, B, c_mod, C, reuse_a, reuse_b)
        acc0 = __builtin_amdgcn_wmma_f32_16x16x4_f32(false, a, false, b0, (short)0, acc0, false, false);
        acc1 = __builtin_amdgcn_wmma_f32_16x16x4_f32(false, a, false, b1, (short)0, acc1, false, false);
        acc2 = __builtin_amdgcn_wmma_f32_16x16x4_f32(false, a, false, b2, (short)0, acc2, false, false);
        acc3 = __builtin_amdgcn_wmma_f32_16x16x4_f32(false, a, false, b3, (short)0, acc3, false, false);
    }

#pragma unroll
    for (int i = 0; i < 8; ++i) {
        const int row = row_tile * 16 + i + ((lane >> 4) << 3);  // M split at lane 16
        float* Crow = C + (size_t)row * FDIM + ncl;
        Crow[0]  = acc0[i];
        Crow[16] = acc1[i];
        Crow[32] = acc2[i];
        Crow[48] = acc3[i];
    }
}

// ---------------- aggregation: self-loop init + edge scatter + relu ----------------

__global__ void agg_init(const float* __restrict__ t, const float* __restrict__ dinv,
                         const float* __restrict__ bias, float* __restrict__ out) {
    size_t i = (size_t)blockIdx.x * blockDim.x + threadIdx.x;
    if (i >= (size_t)NNODES * FDIM) return;
    const int node = (int)(i / FDIM);
    const int f    = (int)(i % FDIM);
    const float dv = dinv[node];
    out[i] = dv * dv * t[i] + bias[f];
}

__global__ void agg_edges(const float* __restrict__ t, const float* __restrict__ dinv,
                          const int* __restrict__ src, const int* __restrict__ dst,
                          float* __restrict__ out) {
    const long long gid = (long long)blockIdx.x * blockDim.x + threadIdx.x;
    const long long e = gid >> 4;          // 16 threads per edge, 4 floats each
    if (e >= NEDGES) return;
    const int f = (int)(gid & 15) * 4;
    const int s = src[e];
    const int d = dst[e];
    const float norm = dinv[s] * dinv[d];
    const float4 v = *(const float4*)(t + (size_t)s * FDIM + f);
    float* o = out + (size_t)d * FDIM + f;
    atomicAdd(o + 0, norm * v.x);
    atomicAdd(o + 1, norm * v.y);
    atomicAdd(o + 2, norm * v.z);
    atomicAdd(o + 3, norm * v.w);
}

__global__ void relu_inplace(float* __restrict__ p) {
    size_t i = (size_t)blockIdx.x * blockDim.x + threadIdx.x;
    if (i < (size_t)NNODES * FDIM) p[i] = fmaxf(p[i], 0.0f);
}

__global__ void copy_f32(const float* __restrict__ a, float* __restrict__ b) {
    size_t i = (size_t)blockIdx.x * blockDim.x + threadIdx.x;
    if (i < (size_t)NNODES * FDIM) b[i] = a[i];
}

// ---------------- host-side orchestration ----------------

extern "C" void kernel_launch(void* const* d_in, const int* in_sizes, int n_in,
                              void* d_out, int out_size, void* d_ws, size_t ws_size,
                              hipStream_t stream) {
    const float* x     = (const float*)d_in[0];
    const int*   ei    = (const int*)  d_in[1];   // [2, E]: src row then dst row
    const float* W0    = (const float*)d_in[4];
    const float* b0    = (const float*)d_in[5];
    const float* Wh    = (const float*)d_in[6];   // [2,64,64]
    const float* bh    = (const float*)d_in[7];   // [2,64]
    const float* bn1_g = (const float*)d_in[8];
    const float* bn1_b = (const float*)d_in[9];
    const float* bn1_m = (const float*)d_in[10];
    const float* bn1_v = (const float*)d_in[11];
    const float* bnh_g = (const float*)d_in[12];  // [2,64]
    const float* bnh_b = (const float*)d_in[13];
    const float* bnh_m = (const float*)d_in[14];
    const float* bnh_v = (const float*)d_in[15];

    const int* src = ei;
    const int* dst = ei + NEDGES;

    float* out  = (float*)d_out;
    float* h1   = out + (size_t)1 * NNODES * FDIM;
    float* h2   = out + (size_t)2 * NNODES * FDIM;
    float* h3   = out + (size_t)3 * NNODES * FDIM;

    float* wsf   = (float*)d_ws;
    float* dinv  = wsf;                        // N floats (deg, then dinv in place)
    float* scl   = wsf + NNODES;               // 64
    float* shf   = wsf + NNODES + 64;          // 64
    float* tbuf  = wsf + NNODES + 128;         // N*64 GEMM output

    const int NF        = NNODES * FDIM;       // 6.4M
    const int nf_blocks = (NF + 255) / 256;
    const int e_blocks  = (NEDGES + 255) / 256;
    const int e16_blocks = (int)(((long long)NEDGES * 16 + 255) / 256);
    const int row_blocks = NNODES / 16;                          // 6250 row tiles
    const int gemm_blocks = (row_blocks + 7) / 8;                // 8 waves/block

    // degrees -> dinv (shared by all 3 layers)
    zero_f32<<<(NNODES + 255) / 256, 256, 0, stream>>>(dinv, NNODES);
    deg_count<<<e_blocks, 256, 0, stream>>>(dst, dinv);
    make_dinv<<<(NNODES + 255) / 256, 256, 0, stream>>>(dinv);

    // one GCN layer: h_out = relu( scatter( (h_in*BN) @ W ) + b )
    auto layer = [&](const float* hin, const float* W, const float* bias,
                     const float* g, const float* bb, const float* m, const float* v,
                     float* hout) {
        bn_prep<<<1, 64, 0, stream>>>(g, bb, m, v, scl, shf);
        bn_gemm_wmma<<<gemm_blocks, 256, 0, stream>>>(hin, W, scl, shf, tbuf, NNODES);
        agg_init<<<nf_blocks, 256, 0, stream>>>(tbuf, dinv, bias, hout);
        agg_edges<<<e16_blocks, 256, 0, stream>>>(tbuf, dinv, src, dst, hout);
        relu_inplace<<<nf_blocks, 256, 0, stream>>>(hout);
    };

    layer(x,  W0,             b0,        bn1_g,      bn1_b,      bn1_m,      bn1_v,      h1);
    layer(h1, Wh,             bh,        bnh_g,      bnh_b,      bnh_m,      bnh_v,      h2);
    layer(h2, Wh + 64 * 64,   bh + 64,   bnh_g + 64, bnh_b + 64, bnh_m + 64, bnh_v + 64, h3);

    // output tuple is (h3, h1, h2, h3): fill slot 0 with h3
    copy_f32<<<nf_blocks, 256, 0, stream>>>(h3, out);
}